// BruteForceIndex_9500467659388
// MI455X (gfx1250) — compile-verified
//
#include <hip/hip_runtime.h>
#include <cfloat>
#include <climits>

// ---------------------------------------------------------------------------
// Brute-force top-K retrieval, fused score + select, fp32 WMMA (gfx1250).
//   B=1024 users, D=128, N=500000 candidates, K=100.
// Phase 1: grid (64 user-tiles x 31 chunks); each block scores 16 users x
//          16384 candidates with V_WMMA_F32_16X16X4_F32 (two independent
//          accumulator chains per wave for ILP) and keeps an exact running
//          top-100 per user in LDS (threshold filter + replace-min).
// Phase 2: one block per user merges 31 partial top-100 lists exactly.
// ---------------------------------------------------------------------------

typedef float v2f __attribute__((ext_vector_type(2)));
typedef float v8f __attribute__((ext_vector_type(8)));

#define B_USERS   1024
#define DIM       128
#define NCAND     500000
#define TOPK      100
#define CHUNK     16384
#define NCHUNKS   31          // ceil(500000/16384); last chunk = 8480
#define UT        16          // users per block
#define SUB       256         // candidates per compute round (2 tiles / wave)
#define HALF      128         // candidates per scan/drain round
#define SSTRIDE   260         // scoreBuf row stride (floats), bank pad
#define USTRIDE   132         // uEmb row stride (floats), bank pad
#define NPART     (NCHUNKS * TOPK)   // 3100 partial entries per user

__global__ __launch_bounds__(256)
void bf_topk_phase1(const float* __restrict__ users,
                    const float* __restrict__ cand,
                    float* __restrict__ wsS, int* __restrict__ wsI) {
  __shared__ float uEmb[UT * USTRIDE];       // 8.4 KB
  __shared__ float scoreBuf[UT * SSTRIDE];   // 16.6 KB
  __shared__ float topS[UT][TOPK];           // 6.4 KB
  __shared__ int   topI[UT][TOPK];           // 6.4 KB
  __shared__ float queueS[UT][HALF];         // 8 KB
  __shared__ int   queueI[UT][HALF];         // 8 KB
  __shared__ int   qcount[UT];
  __shared__ float thresh[UT];
  __shared__ int   minPos[UT];

  const int tid  = threadIdx.x;
  const int lane = tid & 31;
  const int wave = tid >> 5;                 // 0..7
  const int ut   = blockIdx.x;               // user tile (x fast => chunk shared in L2)
  const int ch   = blockIdx.y;               // candidate chunk
  const int userBase  = ut * UT;
  const int chunkBase = ch * CHUNK;
  const int chunkLen  = min(CHUNK, NCAND - chunkBase);
  const int nSub      = (chunkLen + SUB - 1) / SUB;

  // ---- init: stage 16 user rows into LDS, clear top lists ----
  for (int i = tid; i < UT * DIM; i += 256) {
    const int u = i >> 7, d = i & 127;
    uEmb[u * USTRIDE + d] = users[(userBase + u) * DIM + d];
  }
  for (int i = tid; i < UT * TOPK; i += 256) {
    topS[i / TOPK][i % TOPK] = -FLT_MAX;
    topI[i / TOPK][i % TOPK] = 0;
  }
  if (tid < UT) { qcount[tid] = 0; thresh[tid] = -FLT_MAX; minPos[tid] = 0; }
  __syncthreads();

  // ---- hoist A fragments (user matrix, constant for the whole chunk) ----
  // A 16x4 fp32 layout: lanes 0-15 hold (M=lane, K={0,1}); lanes 16-31 (M, K={2,3})
  const int halfSel = (lane >> 4) << 1;      // 0 or 2
  const int uRow    = lane & 15;
  v2f afrag[32];
#pragma unroll
  for (int kk = 0; kk < 32; ++kk) {
    const int d0 = kk * 4 + halfSel;
    afrag[kk].x = uEmb[uRow * USTRIDE + d0];
    afrag[kk].y = uEmb[uRow * USTRIDE + d0 + 1];
  }

  for (int sb = 0; sb < nSub; ++sb) {
    const int subBase = sb * SUB;
    // two 16-candidate tiles per wave -> two independent WMMA chains
    const int nbLoc0 = subBase + wave * 16;
    const int nbLoc1 = subBase + HALF + wave * 16;
    const int nG0 = chunkBase + nbLoc0 + (lane & 15);
    const int nG1 = chunkBase + nbLoc1 + (lane & 15);
    const float* crow0 = cand + (long long)min(nG0, NCAND - 1) * DIM;
    const float* crow1 = cand + (long long)min(nG1, NCAND - 1) * DIM;

    // prefetch next round's rows into near caches (WGP-scope)
    __builtin_prefetch(cand + (long long)min(nG0 + SUB, NCAND - 1) * DIM, 0, 3);
    __builtin_prefetch(cand + (long long)min(nG1 + SUB, NCAND - 1) * DIM, 0, 3);

    v8f c0 = {0.f, 0.f, 0.f, 0.f, 0.f, 0.f, 0.f, 0.f};
    v8f c1 = {0.f, 0.f, 0.f, 0.f, 0.f, 0.f, 0.f, 0.f};
#pragma unroll
    for (int kk = 0; kk < 32; ++kk) {
      const int d0 = kk * 4 + halfSel;
      // B 4x16 fp32: lanes 0-15 hold (K={0,1}, N=lane); lanes 16-31 (K={2,3}, N)
      const v2f b0 = { crow0[d0], crow0[d0 + 1] };     // global_load_b64
      const v2f b1 = { crow1[d0], crow1[d0 + 1] };
      c0 = __builtin_amdgcn_wmma_f32_16x16x4_f32(
               false, afrag[kk], false, b0, (short)0, c0, false, false);
      c1 = __builtin_amdgcn_wmma_f32_16x16x4_f32(
               false, afrag[kk], false, b1, (short)0, c1, false, false);
    }

    // D layout: vgpr i -> (M=i, N=lane) lanes 0-15; (M=8+i, N=lane-16) lanes 16-31
    const int cLoc0 = nbLoc0 - subBase + (lane & 15);  // 0..127
    const int cLoc1 = nbLoc1 - subBase + (lane & 15);  // 128..255
    const int uOff  = (lane < 16) ? 0 : 8;
#pragma unroll
    for (int i = 0; i < 8; ++i) {
      scoreBuf[(i + uOff) * SSTRIDE + cLoc0] = c0[i];
      scoreBuf[(i + uOff) * SSTRIDE + cLoc1] = c1[i];
    }
    __syncthreads();

    // ---- two scan/drain halves (queue capacity HALF is overflow-proof) ----
#pragma unroll
    for (int h = 0; h < 2; ++h) {
      const int hBase = h * HALF;
      {
        const int su = tid >> 4, sl = tid & 15;
        const float th = thresh[su];
#pragma unroll
        for (int j = 0; j < 8; ++j) {
          const int cc  = hBase + sl + j * 16;
          const float s = scoreBuf[su * SSTRIDE + cc];
          const int gid = chunkBase + subBase + cc;
          if (gid < NCAND && s > th) {
            const int p = atomicAdd(&qcount[su], 1);
            queueS[su][p] = s;
            queueI[su][p] = gid;
          }
        }
      }
      __syncthreads();

      // drain: one thread per user, replace-min into top-100
      if ((tid & 15) == 0) {
        const int u = tid >> 4;
        const int cnt = qcount[u];
        int   mp = minPos[u];
        float mv = topS[u][mp];
        for (int e = 0; e < cnt; ++e) {
          const float s = queueS[u][e];
          if (s > mv) {
            topS[u][mp] = s;
            topI[u][mp] = queueI[u][e];
            mv = topS[u][0]; mp = 0;
            for (int j = 1; j < TOPK; ++j) {
              const float v = topS[u][j];
              if (v < mv) { mv = v; mp = j; }
            }
          }
        }
        minPos[u] = mp; thresh[u] = mv; qcount[u] = 0;
      }
      __syncthreads();
    }
  }

  // ---- write exact per-(user,chunk) top-100 partials ----
  for (int i = tid; i < UT * TOPK; i += 256) {
    const int u = i / TOPK, j = i % TOPK;
    const long long off = ((long long)(userBase + u) * NCHUNKS + ch) * TOPK + j;
    wsS[off] = topS[u][j];
    wsI[off] = topI[u][j];
  }
}

__global__ __launch_bounds__(256)
void bf_topk_phase2(const float* __restrict__ wsS, const int* __restrict__ wsI,
                    const int* __restrict__ ids, int* __restrict__ out) {
  __shared__ float s[NPART];                 // 12.4 KB
  __shared__ int   gi[NPART];                // 12.4 KB
  __shared__ float redS[256];
  __shared__ int   redP[256];

  const int u = blockIdx.x, tid = threadIdx.x;
  for (int i = tid; i < NPART; i += 256) {
    s[i]  = wsS[(long long)u * NPART + i];
    gi[i] = wsI[(long long)u * NPART + i];
  }
  __syncthreads();

  for (int k = 0; k < TOPK; ++k) {
    // local argmax (tie-break: smaller candidate index, matches jax top_k)
    float bs = -FLT_MAX; int bp = -1; int bid = INT_MAX;
    for (int i = tid; i < NPART; i += 256) {
      const float v = s[i]; const int g = gi[i];
      if (v > bs || (v == bs && g < bid)) { bs = v; bid = g; bp = i; }
    }
    redS[tid] = bs; redP[tid] = bp;
    __syncthreads();
    for (int off = 128; off > 0; off >>= 1) {
      if (tid < off) {
        const float v1 = redS[tid], v2 = redS[tid + off];
        const int   p1 = redP[tid], p2 = redP[tid + off];
        const int g1 = (p1 >= 0) ? gi[p1] : INT_MAX;
        const int g2 = (p2 >= 0) ? gi[p2] : INT_MAX;
        if (v2 > v1 || (v2 == v1 && g2 < g1)) { redS[tid] = v2; redP[tid] = p2; }
      }
      __syncthreads();
    }
    if (tid == 0) {
      const int bestPos = redP[0];
      out[u * TOPK + k] = ids[gi[bestPos]];
      s[bestPos] = -FLT_MAX;                 // remove selected entry
    }
    __syncthreads();
  }
}

extern "C" void kernel_launch(void* const* d_in, const int* in_sizes, int n_in,
                              void* d_out, int out_size, void* d_ws, size_t ws_size,
                              hipStream_t stream) {
  const float* users = (const float*)d_in[0];   // [1024,128] f32
  const float* cand  = (const float*)d_in[1];   // [500000,128] f32
  const int*   ids   = (const int*)d_in[2];     // [500000] i32
  int* out = (int*)d_out;                       // [1024,100] i32

  float* wsS = (float*)d_ws;                    // 1024*31*100 f32 = 12.7 MB
  int*   wsI = (int*)((char*)d_ws +
                      sizeof(float) * (size_t)B_USERS * NCHUNKS * TOPK);

  dim3 g1(B_USERS / UT, NCHUNKS);               // 64 x 31 blocks
  bf_topk_phase1<<<g1, 256, 0, stream>>>(users, cand, wsS, wsI);
  bf_topk_phase2<<<B_USERS, 256, 0, stream>>>(wsS, wsI, ids, out);
}